// CustomLayerPCEN2_45947560132627
// MI455X (gfx1250) — compile-verified
//
#include <hip/hip_runtime.h>

// PCEN: data [1, F=1024, T=16384] fp32.
//   m_t = 0.5*m_{t-1} + 0.5*x_t   (S = 0.5 fixed in the layer)
//   out = (x * (m+EPS)^-alpha + delta)^r - delta^r
//
// Parallel-scan formulation: kernel decay 0.5^k => 32-sample lookback gives the
// carry to < 2^-32 (below fp32 ulp).  Within a 16-step block the EMA is a
// 16x16 matmul with a lower-triangular Toeplitz matrix of exact powers of two,
// computed with V_WMMA_F32_16X16X4_F32 (full fp32).  Carry between 16-step
// sub-blocks is exact via lane broadcast of column 15.
//
// Stage-in uses GLOBAL_LOAD_ASYNC_TO_LDS_B128 (ASYNCcnt DMA path) when the
// toolchain exposes it; falls back to load+ds_store otherwise.

typedef float v2f __attribute__((ext_vector_type(2)));
typedef float v8f __attribute__((ext_vector_type(8)));
typedef int   v4i __attribute__((ext_vector_type(4)));

#define PCEN_F 1024
#define PCEN_T 16384
#define LB 32                      // lookback samples (0.5^33 below fp32 ulp)
#define WG_TCOLS 512               // T-columns of output per workgroup
#define WAVE_TCOLS 64              // T-columns per wave (8 waves)
#define LDS_COLS (WG_TCOLS + LB)   // 544
#define LDS_STRIDE 548             // pad: 548 % 64 = 36, 36*r mod 64 distinct for r<16
#define EPS 1e-6f

#define GAS(n) __attribute__((address_space(n)))

#if __has_builtin(__builtin_amdgcn_global_load_async_to_lds_b128)
#define PCEN_ASYNC_LDS 1
#else
#define PCEN_ASYNC_LDS 0
#endif

__device__ __forceinline__ void pcen_async_wait_and_barrier() {
#if PCEN_ASYNC_LDS
#if __has_builtin(__builtin_amdgcn_s_wait_asynccnt)
    __builtin_amdgcn_s_wait_asynccnt(0);
#else
    asm volatile("s_wait_asynccnt 0" ::: "memory");
#endif
#endif
    __syncthreads();
}

__launch_bounds__(256, 2)
__global__ void pcen_wmma_kernel(const float* __restrict__ x,
                                 const float* __restrict__ alpha_p,
                                 const float* __restrict__ r_p,
                                 const float* __restrict__ delta_p,
                                 float* __restrict__ out) {
    __shared__ float lds[16 * LDS_STRIDE];

    const int tid = threadIdx.x;
    const int tiles_t = PCEN_T / WG_TCOLS;            // 32
    const int f0 = (blockIdx.x / tiles_t) * 16;
    const int t0 = (blockIdx.x % tiles_t) * WG_TCOLS;

    // ---- Stage [16 x 544] fp32 tile into LDS, coalesced 16B along T ----
    // 544/4 = 136 float4 segments per row; 16*136 = 2176 segments.
    for (int i = tid; i < 16 * (LDS_COLS / 4); i += 256) {
        const int row = i / (LDS_COLS / 4);
        const int seg = i % (LDS_COLS / 4);
        const int gcol = t0 - LB + seg * 4;
        float* ldst = &lds[row * LDS_STRIDE + seg * 4];
        if (gcol >= 0) {
            const float* gsrc = x + (size_t)(f0 + row) * PCEN_T + gcol;
#if PCEN_ASYNC_LDS
            // Lane-wise async DMA: global -> LDS, 16B, no VGPR round-trip.
            __builtin_amdgcn_global_load_async_to_lds_b128(
                (GAS(1) v4i*)gsrc, (GAS(3) v4i*)ldst, 0, 0);
#else
            *(float4*)ldst = *(const float4*)gsrc;
#endif
        } else {
            // t0==0 tiles: zero-pad the lookback window
            *(float4*)ldst = make_float4(0.f, 0.f, 0.f, 0.f);
        }
    }
    pcen_async_wait_and_barrier();

    const float alpha = alpha_p[0];
    const float r     = r_p[0];
    const float delta = delta_p[0];

    const int wv    = tid >> 5;          // wave id 0..7
    const int lane  = tid & 31;
    const int tcol  = lane & 15;         // N index (output timestep within block)
    const int khalf = (lane >> 4) << 1;  // 0 or 2 : K sub-index held by this half-wave
    const int row_a = lane & 15;         // A-matrix row (frequency) for this lane

    // ---- B fragments: Toeplitz scan weights, exact powers of two ----
    // Main block: W[t,k] = 0.5^(t-k+1) for k<=t else 0  (t,k relative to block)
    v2f bmain[4];
    #pragma unroll
    for (int kb = 0; kb < 4; ++kb)
        #pragma unroll
        for (int j = 0; j < 2; ++j) {
            const int K = kb * 4 + khalf + j;
            bmain[kb][j] = (K <= tcol) ? __builtin_ldexpf(1.0f, -(tcol - K + 1)) : 0.0f;
        }
    // Lookback window (32 cols ending just before the wave's first output col):
    // weight(t', Kw) = 0.5^(t' + 33 - Kw), Kw in [0,32)
    v2f blb[8];
    #pragma unroll
    for (int kb = 0; kb < 8; ++kb)
        #pragma unroll
        for (int j = 0; j < 2; ++j) {
            const int Kw = kb * 4 + khalf + j;
            blb[kb][j] = __builtin_ldexpf(1.0f, -(tcol + 33 - Kw));
        }

    const float cfac = __builtin_ldexpf(1.0f, -(tcol + 1)); // 0.5^(t'+1) carry decay
    const bool  r_is_half = (r == 0.5f);
    const float dpr = r_is_half ? __builtin_amdgcn_sqrtf(delta)
                                : __builtin_amdgcn_exp2f(r * __builtin_amdgcn_logf(delta));
    const int srcl = (lane < 16) ? 15 : 31;     // lane holding column 15 for this half

    const int ldsbase = wv * WAVE_TCOLS;        // LDS col of this wave's lookback start

    // ---- First sub-block: 8 lookback WMMAs build the carry-inclusive state ----
    v8f acc = {};
    #pragma unroll
    for (int kb = 0; kb < 8; ++kb) {
        v2f a = *(const v2f*)(&lds[row_a * LDS_STRIDE + ldsbase + kb * 4 + khalf]);
        acc = __builtin_amdgcn_wmma_f32_16x16x4_f32(false, a, false, blb[kb],
                                                    (short)0, acc, false, false);
    }

    // ---- 4 sub-blocks of 16 output timesteps each ----
    #pragma unroll
    for (int tb = 0; tb < WAVE_TCOLS / 16; ++tb) {
        const int cb = ldsbase + LB + tb * 16;  // LDS col of sub-block start
        #pragma unroll
        for (int kb = 0; kb < 4; ++kb) {
            v2f a = *(const v2f*)(&lds[row_a * LDS_STRIDE + cb + kb * 4 + khalf]);
            acc = __builtin_amdgcn_wmma_f32_16x16x4_f32(false, a, false, bmain[kb],
                                                        (short)0, acc, false, false);
        }
        // acc = M for this 16x16 tile.  Emit PCEN + build carry for next block.
        v8f nextc;
        #pragma unroll
        for (int v = 0; v < 8; ++v) {
            const float m = acc[v];
            // carry: m at t'=15 of this block, broadcast to all lanes of the half
            const float mp = __shfl(m, srcl, 32);
            nextc[v] = mp * cfac;

            const int row = v + ((lane >> 4) << 3);             // 0..15
            const float xv = lds[row * LDS_STRIDE + cb + tcol]; // conflict-free
            // (m+eps)^-alpha folded into a multiply: avoids v_div
            const float rmep = __builtin_amdgcn_exp2f(-alpha * __builtin_amdgcn_logf(m + EPS));
            const float y = __builtin_fmaf(xv, rmep, delta);    // y >= delta > 0
            const float pv = r_is_half ? __builtin_amdgcn_sqrtf(y)
                                       : __builtin_amdgcn_exp2f(r * __builtin_amdgcn_logf(y));
            const size_t oidx = (size_t)(f0 + row) * PCEN_T
                              + (size_t)(t0 + wv * WAVE_TCOLS + tb * 16 + tcol);
            out[oidx] = pv - dpr;
        }
        acc = nextc;
    }
}

extern "C" void kernel_launch(void* const* d_in, const int* in_sizes, int n_in,
                              void* d_out, int out_size, void* d_ws, size_t ws_size,
                              hipStream_t stream) {
    const float* data  = (const float*)d_in[0];
    const float* alpha = (const float*)d_in[1];
    const float* r     = (const float*)d_in[2];
    const float* delta = (const float*)d_in[3];
    float* out = (float*)d_out;

    const int blocks = (PCEN_F / 16) * (PCEN_T / WG_TCOLS);  // 64 * 32 = 2048
    pcen_wmma_kernel<<<blocks, 256, 0, stream>>>(data, alpha, r, delta, out);
}